// AttentionAggregation_67095979098786
// MI455X (gfx1250) — compile-verified
//
#include <hip/hip_runtime.h>
#include <hip/hip_bf16.h>

typedef __attribute__((ext_vector_type(2))) float v2f;
typedef __attribute__((ext_vector_type(8))) float v8f;

#define CH 128
#define HD 8

// ---------------------------------------------------------------------------
// Stage 1: per-edge attention logits via WMMA f32 16x16x4.
// One wave (32 lanes) computes a 16(edge) x 16 tile:
//   cols 0..7  = dot(x[src[e]], a_src[h])   (block-diag B1)
//   cols 8..15 = dot(x[dst[e]], a_dst[h])   (block-diag B2)
// alpha[e][h] = C[e][h] + C[e][h+8], then leaky-relu, store, atomicMax segmax.
// B operand is built branchlessly from 4 preloaded v2f register values.
// ---------------------------------------------------------------------------
__global__ __launch_bounds__(256) void gat_alpha_wmma_kernel(
    const float* __restrict__ x,
    const int* __restrict__ src,
    const int* __restrict__ dst,
    const float* __restrict__ att,     // (8, 32): [h][0..15]=a_src, [h][16..31]=a_dst
    float* __restrict__ alpha,         // (E, 8) raw leaky-relu logits
    float* __restrict__ segmax,        // (N, 8) clamped segment max (uint-ordered)
    int nEdges)
{
    const int lane  = threadIdx.x & 31;
    const int wave  = threadIdx.x >> 5;
    const int m     = lane & 15;     // matrix row (edge within tile) / B column
    const int khalf = lane >> 4;     // 0: K+{0,1}, 1: K+{2,3} per ISA A/B layout

    const long long e0 = ((long long)blockIdx.x * 8 + wave) * 16;
    long long eM = e0 + m;
    const int eClamp = (eM < nEdges) ? (int)eM : (nEdges - 1);

    const int rs = src[eClamp];
    const int rd = dst[eClamp];
    const float* __restrict__ xs = x + (long long)rs * CH + khalf * 2;
    const float* __restrict__ xd = x + (long long)rd * CH + khalf * 2;

    // ---- Preload this lane's 8 relevant att values (branch-free loop body) --
    // Lane holds B column n = m.  n<8 -> a_src block, n>=8 -> a_dst block.
    const int  hb       = m & 7;        // head block this column belongs to
    const bool isSrcCol = (m < 8);
    const float* __restrict__ ab =
        att + hb * 32 + (isSrcCol ? 0 : 16) + khalf * 2;
    v2f pre[4];
#pragma unroll
    for (int j = 0; j < 4; ++j)
        pre[j] = *(const v2f*)(ab + j * 4);

    v8f c = {0.f, 0.f, 0.f, 0.f, 0.f, 0.f, 0.f, 0.f};

    for (int kko = 0; kko < 8; ++kko) {          // head-block sized K chunks
        const bool act  = (kko == hb);           // this chunk hits lane's block
        const bool actS = act && isSrcCol;
        const bool actD = act && !isSrcCol;
#pragma unroll
        for (int j = 0; j < 4; ++j) {
            const int kk = kko * 4 + j;          // 4-wide K step index
            // A operands: lane holds (row m, K = 4*kk + 2*khalf + {0,1})
            v2f aS = *(const v2f*)(xs + kk * 4);
            v2f aD = *(const v2f*)(xd + kk * 4);
            // B operands: pure register selects (v_cndmask), no branches
            v2f bS, bD;
            bS.x = actS ? pre[j].x : 0.f;
            bS.y = actS ? pre[j].y : 0.f;
            bD.x = actD ? pre[j].x : 0.f;
            bD.y = actD ? pre[j].y : 0.f;

            c = __builtin_amdgcn_wmma_f32_16x16x4_f32(false, aS, false, bS,
                                                      (short)0, c, false, false);
            c = __builtin_amdgcn_wmma_f32_16x16x4_f32(false, aD, false, bD,
                                                      (short)0, c, false, false);
        }
    }

    // C layout: VGPR i, lane L -> (row = (L<16 ? i : i+8), col = L&15).
    // Fold col h with col h+8 via xor-8 shuffle (same row half).
#pragma unroll
    for (int i = 0; i < 8; ++i) {
        float s = c[i] + __shfl_xor(c[i], 8, 32);
        if ((lane & 15) < 8) {
            const int h = lane & 7;
            const long long e = e0 + i + khalf * 8;
            if (e < nEdges) {
                float lr = (s >= 0.f) ? s : 0.2f * s;   // leaky relu
                alpha[e * HD + h] = lr;
                // max(seg_max, 0) == seg_max over max(alpha, 0) with zero init;
                // non-negative floats order correctly as uint bit patterns.
                float cl = fmaxf(lr, 0.f);
                int dn = dst[e];
                atomicMax((unsigned int*)&segmax[(long long)dn * HD + h],
                          __float_as_uint(cl));
            }
        }
    }
}

// ---------------------------------------------------------------------------
// Stage 2: w = exp(alpha - segmax[dst]); segsum[dst] += w  (one thread / (e,h))
// ---------------------------------------------------------------------------
__global__ __launch_bounds__(256) void gat_expsum_kernel(
    const int* __restrict__ dst,
    const float* __restrict__ segmax,
    float* __restrict__ alpha,         // in: logits, out: exp weights
    float* __restrict__ segsum,
    int nEdges)
{
    int t = blockIdx.x * blockDim.x + threadIdx.x;
    if (t >= nEdges * HD) return;
    const int e = t >> 3;
    const int h = t & 7;
    const int d = dst[e];
    float w = __expf(alpha[t] - segmax[d * HD + h]);
    alpha[t] = w;
    atomicAdd(&segsum[d * HD + h], w);
}

// ---------------------------------------------------------------------------
// Stage 3: out[dst] += (w / max(segsum[dst],1e-10)) * x[src]
// One thread per (edge, 4 channels): float4 gather + 4 float atomics.
// ---------------------------------------------------------------------------
__global__ __launch_bounds__(256) void gat_scatter_kernel(
    const float* __restrict__ x,
    const int* __restrict__ src,
    const int* __restrict__ dst,
    const float* __restrict__ w,
    const float* __restrict__ segsum,
    float* __restrict__ out,
    int nEdges)
{
    long long t = (long long)blockIdx.x * blockDim.x + threadIdx.x;
    if (t >= (long long)nEdges * 32) return;
    const int e  = (int)(t >> 5);
    const int c4 = ((int)t & 31) * 4;
    const int h  = c4 >> 4;

    const int s = src[e];
    const int d = dst[e];
    const float denom = fmaxf(segsum[d * HD + h], 1e-10f);
    const float coeff = w[e * HD + h] / denom;

    const float4 v = *(const float4*)(x + (long long)s * CH + c4);
    float* o = out + (long long)d * CH + c4;
    atomicAdd(o + 0, coeff * v.x);
    atomicAdd(o + 1, coeff * v.y);
    atomicAdd(o + 2, coeff * v.z);
    atomicAdd(o + 3, coeff * v.w);
}

extern "C" void kernel_launch(void* const* d_in, const int* in_sizes, int n_in,
                              void* d_out, int out_size, void* d_ws, size_t ws_size,
                              hipStream_t stream) {
    const float* x         = (const float*)d_in[0];
    const int*   edge_idx  = (const int*)d_in[1];   // (2, E) row-major
    const float* att       = (const float*)d_in[2]; // (8, 32)

    const int E = in_sizes[1] / 2;
    const int N = in_sizes[0] / CH;
    if (E <= 0 || N <= 0) return;

    const int* src = edge_idx;
    const int* dst = edge_idx + E;

    float* alpha  = (float*)d_ws;                       // E*8 floats
    float* segmax = alpha + (size_t)E * HD;             // N*8 floats
    float* segsum = segmax + (size_t)N * HD;            // N*8 floats

    hipMemsetAsync(d_out, 0, (size_t)out_size * sizeof(float), stream);
    hipMemsetAsync(segmax, 0, (size_t)N * HD * sizeof(float), stream);
    hipMemsetAsync(segsum, 0, (size_t)N * HD * sizeof(float), stream);

    // Stage 1: 16 edges per wave, 8 waves per block
    const int tiles   = (E + 15) / 16;
    const int blocksA = (tiles + 7) / 8;
    gat_alpha_wmma_kernel<<<blocksA, 256, 0, stream>>>(x, src, dst, att,
                                                       alpha, segmax, E);

    // Stage 2: one thread per (edge, head)
    const int totB    = E * HD;
    const int blocksB = (totB + 255) / 256;
    gat_expsum_kernel<<<blocksB, 256, 0, stream>>>(dst, segmax, alpha, segsum, E);

    // Stage 3: one thread per (edge, float4-channel-group)
    const long long totC    = (long long)E * 32;
    const int       blocksC = (int)((totC + 255) / 256);
    gat_scatter_kernel<<<blocksC, 256, 0, stream>>>(x, src, dst, alpha, segsum,
                                                    (float*)d_out, E);
}